// DifferentialNetwork_23854248362185
// MI455X (gfx1250) — compile-verified
//
#include <hip/hip_runtime.h>
#include <math.h>

// 4 nets, batch 512, in 12, width 256, out 1.
// One WG = (net n, 8 batch elements). All state in LDS (~312 KB of the 320 KB WGP LDS).
// d_out layout: out[4*512] | dh_out[4*512*12] | d2_out[4*512*144]

typedef float v2f __attribute__((ext_vector_type(2)));
typedef float v8f __attribute__((ext_vector_type(8)));

#define TB      8          // batch elements per WG
#define STRIDE  116        // LDS row stride (floats): bank-conflict-free for B-frag & C-writeback
#define NCOLS   112        // 96 dh cols + 8 t cols + 8 zero-pad
#define NTILES  7          // NCOLS/16

// C = W(256x256, row-major, global) @ in(256 x NCOLS, LDS), written to out(LDS).
// 8 waves: wave w owns M-tile rows {2w, 2w+1} x all 7 N-tiles = 14 C tiles.
__device__ __forceinline__ void gemm256(const float* __restrict__ W,
                                        const float* __restrict__ in,
                                        float* __restrict__ out,
                                        int lane, int wv, bool syncBeforeWrite)
{
    v8f acc[2][NTILES];
    const v8f vz = {0.f, 0.f, 0.f, 0.f, 0.f, 0.f, 0.f, 0.f};
#pragma unroll
    for (int r = 0; r < 2; ++r)
#pragma unroll
        for (int t = 0; t < NTILES; ++t) acc[r][t] = vz;

    const int m0    = (wv * 2 + 0) * 16 + (lane & 15);
    const int m1    = (wv * 2 + 1) * 16 + (lane & 15);
    const int khalf = (lane >> 4) * 2;
    const int nlo   = lane & 15;

    for (int ks = 0; ks < 64; ++ks) {
        const int k = ks * 4 + khalf;
        // A fragments: lane holds W[m][k], W[m][k+1] (8B contiguous, L2-resident)
        v2f a0 = *(const v2f*)(W + m0 * 256 + k);
        v2f a1 = *(const v2f*)(W + m1 * 256 + k);
#pragma unroll
        for (int t = 0; t < NTILES; ++t) {
            const int c = t * 16 + nlo;
            v2f bf;
            bf.x = in[(k + 0) * STRIDE + c];
            bf.y = in[(k + 1) * STRIDE + c];
            acc[0][t] = __builtin_amdgcn_wmma_f32_16x16x4_f32(
                false, a0, false, bf, (short)0, acc[0][t], false, false);
            acc[1][t] = __builtin_amdgcn_wmma_f32_16x16x4_f32(
                false, a1, false, bf, (short)0, acc[1][t], false, false);
        }
    }

    if (syncBeforeWrite) __syncthreads();   // needed when out aliases in

#pragma unroll
    for (int r = 0; r < 2; ++r) {
        const int mbase = (wv * 2 + r) * 16 + 8 * (lane >> 4);
#pragma unroll
        for (int t = 0; t < NTILES; ++t) {
            const int c = t * 16 + nlo;
#pragma unroll
            for (int v = 0; v < 8; ++v)
                out[(mbase + v) * STRIDE + c] = acc[r][t][v];
        }
    }
}

__global__ __launch_bounds__(256, 1)
void diffnet_fused_kernel(const float* __restrict__ x,
                          const float* __restrict__ w0, const float* __restrict__ b0,
                          const float* __restrict__ w1, const float* __restrict__ b1,
                          const float* __restrict__ w2, const float* __restrict__ b2,
                          const float* __restrict__ w3, const float* __restrict__ b3,
                          float* __restrict__ out)
{
    extern __shared__ float sm[];
    float* Abuf = sm;                       // 256*STRIDE  (dh_L / J2hat)
    float* Bbuf = Abuf + 256 * STRIDE;      // 256*STRIDE  (J1hat raw)
    float* gp0  = Bbuf + 256 * STRIDE;      // [256][8] each below
    float* gpp0 = gp0  + 256 * TB;
    float* gp1  = gpp0 + 256 * TB;
    float* gpp1 = gp1  + 256 * TB;
    float* gp2  = gpp1 + 256 * TB;
    float* gpp2 = gp2  + 256 * TB;
    float* s2   = gpp2 + 256 * TB;          // later reused as c2
    float* v2s  = s2   + 256 * TB;          // later reused as c1
    float* v1s  = v2s  + 256 * TB;          // later reused as c0
    float* xl   = v1s  + 256 * TB;          // [8][12]
    float* outa = xl   + TB * 12;           // [8]

    const int tid  = threadIdx.x;
    const int lane = tid & 31;
    const int wv   = tid >> 5;
    const int n    = blockIdx.y;
    const int bg0  = blockIdx.x * TB;       // first batch element of tile

    const float* W0 = w0 + n * 256 * 12;
    const float* W1 = w1 + n * 256 * 256;
    const float* W2 = w2 + n * 256 * 256;
    const float* W3 = w3 + n * 256;         // [1,256] row
    const float* Bi0 = b0 + n * 256;
    const float* Bi1 = b1 + n * 256;
    const float* Bi2 = b2 + n * 256;

    if (tid < TB * 12) xl[tid] = x[bg0 * 12 + tid];
    if (tid < TB)      outa[tid] = b3[n];
    __syncthreads();

    // ---- layer 0 (matvec, N_IN=12) : a0, tanh derivs, dh1 = gp0 (.) W0, t0 cols
    {
        const int i = tid;
        float w0r[12];
#pragma unroll
        for (int m = 0; m < 12; ++m) w0r[m] = W0[i * 12 + m];
        for (int b = 0; b < TB; ++b) {
            float a = Bi0[i];
#pragma unroll
            for (int m = 0; m < 12; ++m) a += w0r[m] * xl[b * 12 + m];
            float t  = tanhf(a);
            float gp = 1.f - t * t;
            gp0[i * TB + b]  = gp;
            gpp0[i * TB + b] = -2.f * t * gp;
#pragma unroll
            for (int m = 0; m < 12; ++m) Abuf[i * STRIDE + b * 12 + m] = gp * w0r[m];
            Abuf[i * STRIDE + 96 + b]  = t;     // t0 column
            Abuf[i * STRIDE + 104 + b] = 0.f;   // zero pad column
        }
    }
    __syncthreads();

    // ---- layer 1 GEMM: Bbuf = W1 @ [dh1 | t0 | 0]  (raw J1hat + raw a1)
    gemm256(W1, Abuf, Bbuf, lane, wv, false);
    __syncthreads();

    // ---- layer 1 activation; Abuf := [dh2 | t1 | 0]; Bbuf (J1hat) preserved for H1
    {
        const int i = tid;
        for (int b = 0; b < TB; ++b) {
            float a  = Bbuf[i * STRIDE + 96 + b] + Bi1[i];
            float t  = tanhf(a);
            float gp = 1.f - t * t;
            gp1[i * TB + b]  = gp;
            gpp1[i * TB + b] = -2.f * t * gp;
#pragma unroll
            for (int m = 0; m < 12; ++m)
                Abuf[i * STRIDE + b * 12 + m] = gp * Bbuf[i * STRIDE + b * 12 + m];
            Abuf[i * STRIDE + 96 + b] = t;      // pad cols stay 0 from layer 0
        }
    }
    __syncthreads();

    // ---- layer 2 GEMM in-place: Abuf := W2 @ Abuf  (raw J2hat + raw a2)
    gemm256(W2, Abuf, Abuf, lane, wv, true);
    __syncthreads();

    // ---- layer 2 activation, scalar output, s2 = w3 (.) gp2
    {
        const int i = tid;
        for (int b = 0; b < TB; ++b) {
            float a  = Abuf[i * STRIDE + 96 + b] + Bi2[i];
            float t  = tanhf(a);
            float gp = 1.f - t * t;
            gp2[i * TB + b]  = gp;
            gpp2[i * TB + b] = -2.f * t * gp;
            s2[i * TB + b]   = W3[i] * gp;
            atomicAdd(&outa[b], W3[i] * t);     // out = w3 . t2 + b3
        }
    }
    __syncthreads();

    // ---- backward: v2 = W2^T s2   (coalesced: lane i reads W2[o*256+i])
    {
        const int i = tid;
        float acc[TB];
#pragma unroll
        for (int b = 0; b < TB; ++b) acc[b] = 0.f;
        for (int o = 0; o < 256; ++o) {
            float w = W2[o * 256 + i];
#pragma unroll
            for (int b = 0; b < TB; ++b) acc[b] += w * s2[o * TB + b];
        }
#pragma unroll
        for (int b = 0; b < TB; ++b) v2s[i * TB + b] = acc[b];
    }
    __syncthreads();

    // ---- backward: v1 = W1^T (v2 (.) gp1)
    {
        const int i = tid;
        float acc[TB];
#pragma unroll
        for (int b = 0; b < TB; ++b) acc[b] = 0.f;
        for (int o = 0; o < 256; ++o) {
            float w = W1[o * 256 + i];
#pragma unroll
            for (int b = 0; b < TB; ++b) acc[b] += w * v2s[o * TB + b] * gp1[o * TB + b];
        }
#pragma unroll
        for (int b = 0; b < TB; ++b) v1s[i * TB + b] = acc[b];
    }
    __syncthreads();

    // ---- dh_out[b,m] = sum_o s2[o,b] * J2hat[o, col(b,m)]   (uses s2 before reuse)
    for (int idx = tid; idx < TB * 12; idx += 256) {
        const int b = idx / 12, m = idx % 12;
        float acc = 0.f;
        for (int o = 0; o < 256; ++o)
            acc += s2[o * TB + b] * Abuf[o * STRIDE + b * 12 + m];
        out[2048 + (n * 512 + bg0 + b) * 12 + m] = acc;
    }
    if (tid < TB) out[n * 512 + bg0 + tid] = outa[tid];
    __syncthreads();

    // ---- curvature weights in place: c2 <- s2, c1 <- v2s, c0 <- v1s
    {
        const int i = tid;
#pragma unroll
        for (int b = 0; b < TB; ++b) {
            s2[i * TB + b]  = W3[i]          * gpp2[i * TB + b];   // c2
            v2s[i * TB + b] = v2s[i * TB + b] * gpp1[i * TB + b];  // c1
            v1s[i * TB + b] = v1s[i * TB + b] * gpp0[i * TB + b];  // c0
        }
    }
    __syncthreads();

    // ---- d2_out[b,j,k] = W0^T c0 W0 + J1hat^T c1 J1hat + J2hat^T c2 J2hat
    for (int idx = tid; idx < TB * 144; idx += 256) {
        const int b = idx / 144, r = idx % 144;
        const int j = r / 12,   k = r % 12;
        float acc = 0.f;
        for (int i = 0; i < 256; ++i) {
            acc += v1s[i * TB + b] * W0[i * 12 + j] * W0[i * 12 + k];
            acc += v2s[i * TB + b] * Bbuf[i * STRIDE + b * 12 + j] * Bbuf[i * STRIDE + b * 12 + k];
            acc += s2[i * TB + b]  * Abuf[i * STRIDE + b * 12 + j] * Abuf[i * STRIDE + b * 12 + k];
        }
        out[26624 + (n * 512 + bg0 + b) * 144 + r] = acc;
    }
}

extern "C" void kernel_launch(void* const* d_in, const int* in_sizes, int n_in,
                              void* d_out, int out_size, void* d_ws, size_t ws_size,
                              hipStream_t stream)
{
    (void)in_sizes; (void)n_in; (void)out_size; (void)d_ws; (void)ws_size;
    const float* x  = (const float*)d_in[0];
    const float* w0 = (const float*)d_in[1];
    const float* b0 = (const float*)d_in[2];
    const float* w1 = (const float*)d_in[3];
    const float* b1 = (const float*)d_in[4];
    const float* w2 = (const float*)d_in[5];
    const float* b2 = (const float*)d_in[6];
    const float* w3 = (const float*)d_in[7];
    const float* b3 = (const float*)d_in[8];
    float* outp = (float*)d_out;

    // LDS: 2 buffers + 9 [256][8] arrays + x tile + out accum
    const size_t smem = (size_t)(2 * 256 * STRIDE + 9 * 256 * TB + TB * 12 + TB) * sizeof(float);
    (void)hipFuncSetAttribute(reinterpret_cast<const void*>(&diffnet_fused_kernel),
                              hipFuncAttributeMaxDynamicSharedMemorySize, (int)smem);

    dim3 grid(512 / TB, 4);   // (batch tiles, nets) = (64, 4)
    dim3 block(256);          // 8 wave32 waves
    diffnet_fused_kernel<<<grid, block, smem, stream>>>(x, w0, b0, w1, b1, w2, b2, w3, b3, outp);
}